// CPTensor_69466801045559
// MI455X (gfx1250) — compile-verified
//
#include <hip/hip_runtime.h>

// CP tensor reconstruction: out[i0, i2, i1] = sum_r f0[i0,r] * f2[i2,r] * f1[i1,r]
// GEMM view: out(512 x 131072) = f0(512 x 32) * KR^T, KR[(i2*512+i1), r] = f2[i2,r]*f1[i1,r]
// Write-bandwidth-bound (256 MB out); V_WMMA_F32_16X16X4_F32 keeps f32 accuracy
// while exceeding the matrix-op rate needed to saturate 23.3 TB/s HBM writes.

typedef float v2f __attribute__((ext_vector_type(2)));
typedef float v8f __attribute__((ext_vector_type(8)));

#define N0   512
#define N1   512
#define N2   256
#define RK   32
#define NCOL (N1 * N2)   // 131072 output columns

__global__ __launch_bounds__(256) void cp_recover_wmma_f32(
    const float* __restrict__ f0,
    const float* __restrict__ f1,
    const float* __restrict__ f2,
    float* __restrict__ out)
{
    const int lane = threadIdx.x & 31;   // wave32
    const int wave = threadIdx.x >> 5;   // 0..7
    const int half = lane >> 4;          // 0 or 1 (half-wave)
    const int lm   = lane & 15;

    // Column tile: 16 consecutive j = i2*512 + i1 values (one i2, 16 i1's)
    const int jt  = blockIdx.x;          // 0..8191
    const int i2  = jt >> 5;             // f2 row
    const int i1b = (jt & 31) << 4;      // base of 16 f1 rows
    const long long jb = (long long)i2 * N1 + i1b;

    // ---- Build B = KR tile (32x16, K-major slices of 4) once per wave.
    // 32-bit B 4x16 layout, slice s: VGPR0 holds k=4s+2h (lanes h*16..h*16+15),
    // VGPR1 holds k=4s+2h+1; column n = lane&15.
    v2f B[8];
    {
        const float* f1row = f1 + (size_t)(i1b + lm) * RK;
        const float* f2row = f2 + (size_t)i2 * RK;
#pragma unroll
        for (int s = 0; s < 8; ++s) {
            const int k = 4 * s + 2 * half;
            const v2f b1 = *(const v2f*)(f1row + k);
            const v2f b2 = *(const v2f*)(f2row + k);
            v2f b; b.x = b1.x * b2.x; b.y = b1.y * b2.y;
            B[s] = b;
        }
    }

    // ---- Each wave covers 4 of the 32 row-tiles (16 rows each) => all 512 rows.
#pragma unroll
    for (int t = 0; t < 4; ++t) {
        const int i0b = (wave + 8 * t) << 4;

        // A tile (16x4 per slice): lane (h*16+m) slice s holds f0[i0b+m, 4s+2h .. +1]
        const float* f0row = f0 + (size_t)(i0b + lm) * RK;
        v2f A[8];
#pragma unroll
        for (int s = 0; s < 8; ++s) {
            const int k = 4 * s + 2 * half;
            A[s] = *(const v2f*)(f0row + k);
        }

        // 8 chained f32 WMMAs accumulate K=32
        v8f c = {};
#pragma unroll
        for (int s = 0; s < 8; ++s) {
            c = __builtin_amdgcn_wmma_f32_16x16x4_f32(
                    /*neg_a=*/false, A[s],
                    /*neg_b=*/false, B[s],
                    /*c_mod=*/(short)0, c,
                    /*reuse_a=*/false, /*reuse_b=*/false);
        }

        // D layout: VGPR d holds row M = d + 8*half, col N = lane&15.
        // Base address per lane; per-d offset d*NCOL*4 = d*512KB fits imm24.
        float* outp = out + ((size_t)(i0b + 8 * half) * NCOL + (size_t)jb + lm);
#pragma unroll
        for (int d = 0; d < 8; ++d) {
            outp[(size_t)d * NCOL] = c[d];
        }
    }
}

extern "C" void kernel_launch(void* const* d_in, const int* in_sizes, int n_in,
                              void* d_out, int out_size, void* d_ws, size_t ws_size,
                              hipStream_t stream) {
    const float* f0 = (const float*)d_in[0];   // (512, 32)
    const float* f1 = (const float*)d_in[1];   // (512, 32)
    const float* f2 = (const float*)d_in[2];   // (256, 32)
    float* out = (float*)d_out;                // 512*256*512 floats

    const int num_col_tiles = NCOL / 16;       // 8192 blocks
    cp_recover_wmma_f32<<<num_col_tiles, 256, 0, stream>>>(f0, f1, f2, out);
}